// DeformableSampler_39779987095812
// MI455X (gfx1250) — compile-verified
//
#include <hip/hip_runtime.h>
#include <hip/hip_bf16.h>

typedef __attribute__((ext_vector_type(2))) float v2f;

#define Bdim 8
#define Cdim 64
#define Hdim 160
#define Wdim 160
#define Kdim 4
static constexpr int HW = Hdim * Wdim;

// saddr-form 8-byte gather with CDNA5 temporal hint: NT in WGP near-cache
// (no reuse there), RT in far cache (x lives resident in the 192MB L2).
__device__ __forceinline__ v2f gather_nt(unsigned voff, const float* sbase, int ioff) {
  v2f r;
  asm volatile("global_load_b64 %0, %1, %2 offset:%3 th:TH_LOAD_NT_RT"
               : "=v"(r)
               : "v"(voff), "s"(sbase), "n"(ioff));
  return r;
}

__global__ __launch_bounds__(256)
void DeformableSampler_39779987095812_kernel(const float* __restrict__ x,
                                             const float* __restrict__ offsets,
                                             const float* __restrict__ scales,
                                             float* __restrict__ out) {
  extern __shared__ float tile[];                 // 64 x 9 floats (padded, bank-conflict free)

  const int lane = threadIdx.x & 31;
  const int wv   = threadIdx.x >> 5;              // wave id == column within block's 8-pixel strip
  // One wave == one spatial position (b,h,w). Force wave-uniform so all
  // coordinate math lands in SALU/SGPRs and offset loads become scalar loads.
  const int pos = __builtin_amdgcn_readfirstlane((int)(blockIdx.x * 8u + wv));
  const int b  = pos / HW;
  const int hw = pos - b * HW;
  const int h  = hw / Wdim;
  const int w  = hw - h * Wdim;

  const float bx = -1.0f + 2.0f * (float)w * (1.0f / (float)(Wdim - 1));
  const float by = -1.0f + 2.0f * (float)h * (1.0f / (float)(Hdim - 1));

  const int offBase = b * (2 * Kdim) * HW + hw;   // offsets[B, 2K, H, W]
  const int sclBase = b * Kdim * HW + hw;         // scales [B, K, H, W]

  const float* xb = x + (size_t)b * Cdim * HW;    // this batch's image stack

  float wx[Kdim], wy[Kdim], ak[Kdim];
  const float* basek[Kdim];                       // wave-uniform row base per k

  #pragma unroll
  for (int k = 0; k < Kdim; ++k) {
    const float ox = offsets[offBase + (2 * k + 0) * HW];
    const float oy = offsets[offBase + (2 * k + 1) * HW];
    const float s  = scales[sclBase + k * HW];
    const float gx = bx + ox;
    const float gy = by + oy;
    // align_corners=True, padding border
    float ix = fminf(fmaxf((gx + 1.0f) * (0.5f * (float)(Wdim - 1)), 0.0f), (float)(Wdim - 1));
    float iy = fminf(fmaxf((gy + 1.0f) * (0.5f * (float)(Hdim - 1)), 0.0f), (float)(Hdim - 1));
    int x0 = (int)ix; if (x0 > Wdim - 2) x0 = Wdim - 2;   // border fold: wx->1 reproduces clamp
    int y0 = (int)iy; if (y0 > Hdim - 2) y0 = Hdim - 2;
    wx[k] = ix - (float)x0;
    wy[k] = iy - (float)y0;
    basek[k] = xb + (y0 * Wdim + x0);
    ak[k] = 0.25f / (1.0f + __expf(-s));          // sigmoid * (1/K)
  }

  // lane covers channels {lane, lane+32}; per-lane byte offsets into the stack
  const unsigned voff0 = (unsigned)lane * (unsigned)(HW * 4);
  const unsigned voff1 = voff0 + (unsigned)(32 * HW * 4);

  // Issue all 16 gathers (4 k * 2 channels * 2 bilinear rows), then one wait.
  v2f v[16];
  #pragma unroll
  for (int k = 0; k < Kdim; ++k) {
    v[k * 4 + 0] = gather_nt(voff0, basek[k], 0);
    v[k * 4 + 1] = gather_nt(voff0, basek[k], Wdim * 4);   // next row, +640B
    v[k * 4 + 2] = gather_nt(voff1, basek[k], 0);
    v[k * 4 + 3] = gather_nt(voff1, basek[k], Wdim * 4);
  }
  asm volatile("s_wait_loadcnt 0"
               : "+v"(v[0]), "+v"(v[1]), "+v"(v[2]),  "+v"(v[3]),
                 "+v"(v[4]), "+v"(v[5]), "+v"(v[6]),  "+v"(v[7]),
                 "+v"(v[8]), "+v"(v[9]), "+v"(v[10]), "+v"(v[11]),
                 "+v"(v[12]),"+v"(v[13]),"+v"(v[14]), "+v"(v[15]));

  float acc0 = 0.0f, acc1 = 0.0f;
  #pragma unroll
  for (int k = 0; k < Kdim; ++k) {
    {
      const v2f t = v[k * 4 + 0], bo = v[k * 4 + 1];
      const float top = t.x + wx[k] * (t.y - t.x);
      const float bot = bo.x + wx[k] * (bo.y - bo.x);
      acc0 += ak[k] * (top + wy[k] * (bot - top));
    }
    {
      const v2f t = v[k * 4 + 2], bo = v[k * 4 + 3];
      const float top = t.x + wx[k] * (t.y - t.x);
      const float bot = bo.x + wx[k] * (bo.y - bo.x);
      acc1 += ak[k] * (top + wy[k] * (bot - top));
    }
  }

  // ---- Transpose through LDS, then block-coalesced async stores from LDS ----
  // tile[c][col], row stride 9 (coprime with 64 banks -> conflict-free)
  tile[lane * 9 + wv]        = acc0;
  tile[(lane + 32) * 9 + wv] = acc1;
  __syncthreads();

  // thread t moves 8B of channel-row c: out[b, c, h, w0 + j .. j+1]
  const int t = (int)threadIdx.x;
  const int c = t >> 2;
  const int j = (t & 3) * 2;
  const float* outBase = out + ((size_t)b * Cdim * HW + (size_t)h * Wdim + (size_t)(w - wv));
  const unsigned gvoff   = (unsigned)(c * HW + j) * 4u;
  const unsigned ldsaddr = (unsigned)(uintptr_t)(&tile[c * 9 + j]);   // flat-LDS low 32 bits

  asm volatile("global_store_async_from_lds_b64 %0, %1, %2"
               :
               : "v"(gvoff), "v"(ldsaddr), "s"(outBase)
               : "memory");
  asm volatile("s_wait_asynccnt 0" ::: "memory");
}

extern "C" void kernel_launch(void* const* d_in, const int* in_sizes, int n_in,
                              void* d_out, int out_size, void* d_ws, size_t ws_size,
                              hipStream_t stream) {
  const float* x       = (const float*)d_in[0];
  const float* offsets = (const float*)d_in[1];
  const float* scales  = (const float*)d_in[2];
  float* out = (float*)d_out;

  // 204,800 positions, one wave each, 8 waves (256 threads) per block;
  // a block covers 8 consecutive w of one (b,h) row (8 | W).
  const int total_pos = Bdim * Hdim * Wdim;
  const int blocks = total_pos / 8;            // 25,600 exactly
  const size_t smem = (size_t)Cdim * 9 * sizeof(float);   // 2304 B
  DeformableSampler_39779987095812_kernel<<<blocks, 256, smem, stream>>>(x, offsets, scales, out);
}